// appnp_3667902071138
// MI455X (gfx1250) — compile-verified
//
#include <hip/hip_runtime.h>
#include <hip/hip_bf16.h>

typedef __attribute__((ext_vector_type(2))) float v2f;
typedef __attribute__((ext_vector_type(4))) float v4f;
typedef __attribute__((ext_vector_type(8))) float v8f;

#define NN 100000
#define EE 3200000
#define FIN 512
#define FH 128
#define FO 40
#define FOQ 10      // FO / 4 float4 quads per row
#define FOP 48      // FO padded to multiple of 16
#define KSTEPS 20
#define ALPHA_F 0.1f
#define OMA_F 0.9f   // 1 - alpha

// ---------------- degree / normalization ----------------

__global__ void deg_init_k(float* __restrict__ deg, int n) {
    int i = blockIdx.x * blockDim.x + threadIdx.x;
    if (i < n) deg[i] = 1.0f;   // self-loop contributes 1 to every node's degree
}

__global__ void deg_count_k(const int* __restrict__ col, float* __restrict__ deg, int e) {
    int i = blockIdx.x * blockDim.x + threadIdx.x;
    if (i < e) atomicAdd(&deg[col[i]], 1.0f);
}

__global__ void dinv_k(float* __restrict__ deg, int n) {
    int i = blockIdx.x * blockDim.x + threadIdx.x;
    if (i < n) deg[i] = rsqrtf(deg[i]);   // deg > 0 always (self-loops)
}

__global__ void edge_w_k(const int* __restrict__ row, const int* __restrict__ col,
                         const float* __restrict__ dinv, float* __restrict__ wedge, int e) {
    int i = blockIdx.x * blockDim.x + threadIdx.x;
    if (i < e) wedge[i] = OMA_F * dinv[row[i]] * dinv[col[i]];
}

// ---------------- pad W2/b2 to 48 columns ----------------

__global__ void pad_w2_k(const float* __restrict__ W2, const float* __restrict__ b2,
                         float* __restrict__ W2p, float* __restrict__ b2p) {
    int i = blockIdx.x * blockDim.x + threadIdx.x;
    if (i < FH * FOP) {
        int r = i / FOP, c = i % FOP;
        W2p[i] = (c < FO) ? W2[r * FO + c] : 0.0f;
    } else if (i < FH * FOP + FOP) {
        int c = i - FH * FOP;
        b2p[c] = (c < FO) ? b2[c] : 0.0f;
    }
}

// ---------------- GEMM1: h1 = relu(x @ W1 + b1), fp32 WMMA ----------------
// block = 256 threads = 8 waves. Block handles 16 rows x 128 cols.
// A tile (16x512 f32 = 32KB) staged in LDS, each wave owns one 16-col slab.

__global__ void __launch_bounds__(256)
gemm1_wmma_k(const float* __restrict__ x, const float* __restrict__ W1,
             const float* __restrict__ b1, float* __restrict__ h1) {
    __shared__ float As[16 * FIN];   // 32 KB
    const int tid = threadIdx.x;
    const int m0 = blockIdx.x * 16;

    for (int i = tid; i < 16 * FIN; i += 256) {
        int r = i >> 9, c = i & (FIN - 1);
        As[i] = x[(size_t)(m0 + r) * FIN + c];
    }
    __syncthreads();

    const int wave = tid >> 5;
    const int lane = tid & 31;
    const int n0 = wave * 16;
    const int m = lane & 15;        // A row within tile (lanes 0-15 / 16-31 both map M=0..15)
    const int n = lane & 15;        // B/C column within tile
    const int half = lane >> 4;     // selects K pair per ISA 16x4 f32 layout
    const int kb = half * 2;

    v8f acc = {};
    for (int k0 = 0; k0 < FIN; k0 += 4) {
        v2f a, b;
        a.x = As[m * FIN + k0 + kb];
        a.y = As[m * FIN + k0 + kb + 1];
        b.x = W1[(size_t)(k0 + kb) * FH + n0 + n];
        b.y = W1[(size_t)(k0 + kb + 1) * FH + n0 + n];
        acc = __builtin_amdgcn_wmma_f32_16x16x4_f32(
            /*neg_a=*/false, a, /*neg_b=*/false, b,
            /*c_mod=*/(short)0, acc, /*reuse_a=*/false, /*reuse_b=*/false);
    }

    const float bias = b1[n0 + n];
#pragma unroll
    for (int v = 0; v < 8; ++v) {
        int rowm = m0 + v + half * 8;                 // C/D layout: M = v + 8*(lane>=16)
        float r = acc[v] + bias;
        h1[(size_t)rowm * FH + n0 + n] = fmaxf(r, 0.0f);
    }
}

// ---------------- GEMM2: h0 = relu(h1 @ W2p + b2p), fp32 WMMA ----------------
// one wave per 16x16 output tile; grid = (6250 m-tiles, 3 n-tiles)

__global__ void __launch_bounds__(32)
gemm2_wmma_k(const float* __restrict__ h1, const float* __restrict__ W2p,
             const float* __restrict__ b2p, float* __restrict__ h0) {
    const int lane = threadIdx.x;
    const int m0 = blockIdx.x * 16;
    const int n0 = blockIdx.y * 16;
    const int m = lane & 15;
    const int n = lane & 15;
    const int half = lane >> 4;
    const int kb = half * 2;

    v8f acc = {};
    for (int k0 = 0; k0 < FH; k0 += 4) {
        v2f a, b;
        a.x = h1[(size_t)(m0 + m) * FH + k0 + kb];
        a.y = h1[(size_t)(m0 + m) * FH + k0 + kb + 1];
        b.x = W2p[(k0 + kb) * FOP + n0 + n];
        b.y = W2p[(k0 + kb + 1) * FOP + n0 + n];
        acc = __builtin_amdgcn_wmma_f32_16x16x4_f32(
            false, a, false, b, (short)0, acc, false, false);
    }

    const int colo = n0 + n;
    if (colo < FO) {                       // drop the padded columns
        const float bias = b2p[colo];
#pragma unroll
        for (int v = 0; v < 8; ++v) {
            int rowm = m0 + v + half * 8;
            float r = acc[v] + bias;
            h0[(size_t)rowm * FO + colo] = fmaxf(r, 0.0f);
        }
    }
}

// ---------------- propagation step ----------------
// zdst = 0.9 * dinv[i]^2 * zsrc   (self-loop edge)  + 0.1 * h0   (teleport)
// one thread per (node, feature-quad); rows are 160B so quads are 16B-aligned

__global__ void prop_init_k(const float* __restrict__ zsrc, const float* __restrict__ h0,
                            const float* __restrict__ dinv, float* __restrict__ zdst, int total) {
    int i = blockIdx.x * blockDim.x + threadIdx.x;   // total = N * FOQ
    if (i < total) {
        int node = i / FOQ;
        float d = dinv[node];
        float s = OMA_F * d * d;
        v4f z = *(const v4f*)(zsrc + (size_t)i * 4);
        v4f h = *(const v4f*)(h0 + (size_t)i * 4);
        v4f o;
        o.x = s * z.x + ALPHA_F * h.x;
        o.y = s * z.y + ALPHA_F * h.y;
        o.z = s * z.z + ALPHA_F * h.z;
        o.w = s * z.w + ALPHA_F * h.w;
        *(v4f*)(zdst + (size_t)i * 4) = o;
    }
}

// zdst[col] += w[e] * zsrc[row] : one thread per (edge, feature-quad).
// float4 gather (global_load_b128) + 4 clause-grouped global_atomic_add_f32;
// row/col/wedge read 10x per edge instead of 40x.

__global__ void prop_edge_k(const int* __restrict__ row, const int* __restrict__ col,
                            const float* __restrict__ wedge,
                            const float* __restrict__ zsrc, float* __restrict__ zdst,
                            int total) {                       // total = E * FOQ (32M, fits int)
    int i = blockIdx.x * blockDim.x + threadIdx.x;
    if (i < total) {
        int e = i / FOQ;
        int q = i - e * FOQ;
        int r = row[e];
        int c = col[e];
        float w = wedge[e];
        v4f z = *(const v4f*)(zsrc + (size_t)r * FO + q * 4);
        float* dst = zdst + (size_t)c * FO + q * 4;
        atomicAdd(dst + 0, w * z.x);
        atomicAdd(dst + 1, w * z.y);
        atomicAdd(dst + 2, w * z.z);
        atomicAdd(dst + 3, w * z.w);
    }
}

// ---------------- host orchestration ----------------

extern "C" void kernel_launch(void* const* d_in, const int* in_sizes, int n_in,
                              void* d_out, int out_size, void* d_ws, size_t ws_size,
                              hipStream_t stream) {
    const float* x  = (const float*)d_in[0];
    const int*   ei = (const int*)d_in[1];
    const float* W1 = (const float*)d_in[2];
    const float* b1 = (const float*)d_in[3];
    const float* W2 = (const float*)d_in[4];
    const float* b2 = (const float*)d_in[5];

    const int N = in_sizes[0] / FIN;      // 100000
    const int E = in_sizes[1] / 2;        // 3200000
    const int* row = ei;                  // edge_index[0] = source
    const int* col = ei + E;              // edge_index[1] = target

    // workspace carve-out (256B aligned slabs)
    char* w = (char*)d_ws;
    auto alloc = [&](size_t bytes) {
        void* p = (void*)w;
        w += (bytes + 255) & ~(size_t)255;
        return p;
    };
    float* dinv  = (float*)alloc((size_t)N * 4);            // deg, then in-place rsqrt
    float* h1    = (float*)alloc((size_t)N * FH * 4);       // 51.2 MB
    float* h0    = (float*)alloc((size_t)N * FO * 4);       // 16 MB
    float* zA    = (float*)alloc((size_t)N * FO * 4);       // 16 MB ping buffer
    float* W2p   = (float*)alloc((size_t)FH * FOP * 4);
    float* b2p   = (float*)alloc((size_t)FOP * 4);
    float* wedge = (float*)alloc((size_t)E * 4);            // 12.8 MB

    const int T = 256;

    // normalization
    deg_init_k<<<(N + T - 1) / T, T, 0, stream>>>(dinv, N);
    deg_count_k<<<(E + T - 1) / T, T, 0, stream>>>(col, dinv, E);
    dinv_k<<<(N + T - 1) / T, T, 0, stream>>>(dinv, N);
    edge_w_k<<<(E + T - 1) / T, T, 0, stream>>>(row, col, dinv, wedge, E);

    // encoder
    pad_w2_k<<<(FH * FOP + FOP + T - 1) / T, T, 0, stream>>>(W2, b2, W2p, b2p);
    gemm1_wmma_k<<<N / 16, 256, 0, stream>>>(x, W1, b1, h1);
    gemm2_wmma_k<<<dim3(N / 16, FOP / 16), 32, 0, stream>>>(h1, W2p, b2p, h0);

    // K=20 propagation steps; even step count -> final write lands in d_out
    const int etot = E * FOQ;             // 32,000,000
    const int ntot = N * FOQ;             // 1,000,000
    const float* zsrc = h0;
    for (int s = 1; s <= KSTEPS; ++s) {
        float* zdst = (s & 1) ? zA : (float*)d_out;
        prop_init_k<<<(ntot + T - 1) / T, T, 0, stream>>>(zsrc, h0, dinv, zdst, ntot);
        prop_edge_k<<<(etot + T - 1) / T, T, 0, stream>>>(row, col, wedge, zsrc, zdst, etot);
        zsrc = zdst;
    }
}